// GINConvNet_17832704213197
// MI455X (gfx1250) — compile-verified
//
#include <hip/hip_runtime.h>
#include <hip/hip_bf16.h>

typedef __attribute__((ext_vector_type(2))) float v2f;
typedef __attribute__((ext_vector_type(8))) float v8f;

#define N_NODES  50000
#define N_EDGES  800000
#define N_GRAPHS 1024
#define BN_EPS   1e-5f

// ---------------------------------------------------------------- elementwise
__global__ void k_copy_f32x4(float4* __restrict__ dst, const float4* __restrict__ src,
                             unsigned n4) {
    unsigned stride = gridDim.x * blockDim.x;
    for (unsigned i = blockIdx.x * blockDim.x + threadIdx.x; i < n4; i += stride)
        dst[i] = src[i];
}

__global__ void k_zero_f32(float* __restrict__ dst, unsigned n) {
    unsigned stride = gridDim.x * blockDim.x;
    for (unsigned i = blockIdx.x * blockDim.x + threadIdx.x; i < n; i += stride)
        dst[i] = 0.0f;
}

// ------------------------------------------------- edge scatter-add (gather + atomic)
// acc[dst[e]*F + f] += feat[src[e]*F + f], two floats per thread (F even).
// Compile-time F -> div/mod become mul-shift; coalesced 8B gathers; f32 atomics.
template<int F>
__global__ void k_scatter_add(float* __restrict__ acc, const float* __restrict__ feat,
                              const int* __restrict__ src, const int* __restrict__ dst,
                              unsigned E) {
    constexpr int F2 = F / 2;
    unsigned total  = E * (unsigned)F2;
    unsigned stride = gridDim.x * blockDim.x;
    for (unsigned i = blockIdx.x * blockDim.x + threadIdx.x; i < total; i += stride) {
        unsigned e  = i / F2;
        unsigned f2 = i - e * F2;
        const float2 v = *(const float2*)(feat + (size_t)src[e] * F + 2 * f2);
        float* p = acc + (size_t)dst[e] * F + 2 * f2;
        atomicAdd(p + 0, v.x);
        atomicAdd(p + 1, v.y);
    }
}

// ---------------------------------------------------------- WMMA f32 GEMM
// out[M x N] = act(A[M x K] @ W[K x N] + bias).
// One wave per 16x32 tile: shared A fragment, two independent WMMA acc chains.
// Main K loop: unconditional clamped-address loads (OOB rows/cols only feed
// never-stored outputs); only the K%4 tail needs value-masking.
__global__ void k_gemm_wmma_f32(const float* __restrict__ A,
                                const float* __restrict__ W,
                                const float* __restrict__ bias,
                                float* __restrict__ out,
                                int M, int K, int N, int doRelu) {
    unsigned tid  = blockIdx.x * blockDim.x + threadIdx.x;
    unsigned wave = tid >> 5;
    unsigned lane = tid & 31u;
    unsigned tilesN  = (unsigned)(N + 15) >> 4;
    unsigned tilesM  = (unsigned)(M + 15) >> 4;
    unsigned groupsN = (tilesN + 1) >> 1;          // 2 N-tiles per wave
    if (wave >= tilesM * groupsN) return;          // wave-uniform: EXEC stays full
    unsigned tm = wave / groupsN;
    unsigned tg = wave - tm * groupsN;

    unsigned half = lane >> 4;                     // 0: K {0,1}, 1: K {2,3}
    unsigned l16  = lane & 15u;

    int row   = (int)(tm * 16 + l16);
    int rowC  = row < M ? row : (M - 1);
    int col0  = (int)(tg * 32 + l16);
    int col1  = col0 + 16;
    int col0C = col0 < N ? col0 : (N - 1);
    int col1C = col1 < N ? col1 : (N - 1);

    const float* pA  = A + (size_t)rowC * K + 2 * half;     // 8B-aligned (K even)
    const float* pB0 = W + (size_t)(2 * half) * N + col0C;
    const float* pB1 = W + (size_t)(2 * half) * N + col1C;
    const size_t stepB = 4 * (size_t)N;

    v8f acc0 = {}, acc1 = {};
    const int K4 = K & ~3;
#pragma unroll 2
    for (int k = 0; k < K4; k += 4) {
        v2f a = *(const v2f*)pA;
        v2f b0, b1;
        b0.x = pB0[0]; b0.y = pB0[N];
        b1.x = pB1[0]; b1.y = pB1[N];
        acc0 = __builtin_amdgcn_wmma_f32_16x16x4_f32(false, a, false, b0,
                                                     (short)0, acc0, false, false);
        acc1 = __builtin_amdgcn_wmma_f32_16x16x4_f32(false, a, false, b1,
                                                     (short)0, acc1, false, false);
        pA += 4; pB0 += stepB; pB1 += stepB;
    }
    if (K4 < K) {                                   // K tail (rem 1..3)
        int k0 = K4 + 2 * (int)half;
        int k1 = k0 + 1;
        int t0 = k0 < K ? k0 : (K - 1);             // clamped, unconditional loads
        int t1 = k1 < K ? k1 : (K - 1);
        const float* ar = A + (size_t)rowC * K;
        const float* w0 = W + (size_t)t0 * N;
        const float* w1 = W + (size_t)t1 * N;
        float a0 = ar[t0], a1 = ar[t1];
        float b00 = w0[col0C], b01 = w1[col0C];
        float b10 = w0[col1C], b11 = w1[col1C];
        v2f a, b0, b1;                              // value-mask after load (cndmask)
        a.x  = k0 < K ? a0  : 0.0f;  a.y  = k1 < K ? a1  : 0.0f;
        b0.x = k0 < K ? b00 : 0.0f;  b0.y = k1 < K ? b01 : 0.0f;
        b1.x = k0 < K ? b10 : 0.0f;  b1.y = k1 < K ? b11 : 0.0f;
        acc0 = __builtin_amdgcn_wmma_f32_16x16x4_f32(false, a, false, b0,
                                                     (short)0, acc0, false, false);
        acc1 = __builtin_amdgcn_wmma_f32_16x16x4_f32(false, a, false, b1,
                                                     (short)0, acc1, false, false);
    }

    float bv0 = bias[col0C];
    float bv1 = bias[col1C];
#pragma unroll
    for (int r = 0; r < 8; ++r) {
        int m = (int)(tm * 16) + r + 8 * (int)half; // C/D: M = r + 8*(lane>=16)
        if (m < M) {
            float* o = out + (size_t)m * N;
            if (col0 < N) {
                float v = acc0[r] + bv0;
                if (doRelu) v = fmaxf(v, 0.0f);
                o[col0] = v;
            }
            if (col1 < N) {
                float v = acc1[r] + bv1;
                if (doRelu) v = fmaxf(v, 0.0f);
                o[col1] = v;
            }
        }
    }
}

// --------------------------------------------------------------- batchnorm
template<int F>
__global__ void k_bn_stats(const float* __restrict__ h, float* __restrict__ stats,
                           unsigned n) {
    __shared__ float sm[2 * F];
    for (unsigned c = threadIdx.x; c < 2 * F; c += blockDim.x) sm[c] = 0.0f;
    __syncthreads();
    unsigned total  = n * (unsigned)F;
    unsigned stride = gridDim.x * blockDim.x;
    for (unsigned i = blockIdx.x * blockDim.x + threadIdx.x; i < total; i += stride) {
        float v = h[i];
        unsigned c = i % (unsigned)F;
        atomicAdd(&sm[c], v);
        atomicAdd(&sm[F + c], v * v);
    }
    __syncthreads();
    for (unsigned c = threadIdx.x; c < 2 * F; c += blockDim.x)
        atomicAdd(&stats[c], sm[c]);
}

template<int F>
__global__ void k_bn_apply(float* __restrict__ h, const float* __restrict__ stats,
                           const float* __restrict__ gamma, const float* __restrict__ beta,
                           unsigned n) {
    unsigned total  = n * (unsigned)F;
    unsigned stride = gridDim.x * blockDim.x;
    const float inv_n = 1.0f / (float)n;
    for (unsigned i = blockIdx.x * blockDim.x + threadIdx.x; i < total; i += stride) {
        unsigned c = i % (unsigned)F;
        float mean = stats[c] * inv_n;
        float var  = stats[F + c] * inv_n - mean * mean;    // biased, as torch
        h[i] = (h[i] - mean) * rsqrtf(var + BN_EPS) * gamma[c] + beta[c];
    }
}

// --------------------------------------------------------------- pooling + head
template<int F>
__global__ void k_pool(const float* __restrict__ h, const int* __restrict__ batch,
                       float* __restrict__ g, unsigned n) {
    unsigned total  = n * (unsigned)F;
    unsigned stride = gridDim.x * blockDim.x;
    for (unsigned i = blockIdx.x * blockDim.x + threadIdx.x; i < total; i += stride) {
        unsigned node = i / (unsigned)F;
        unsigned c    = i - node * (unsigned)F;
        atomicAdd(&g[(size_t)batch[node] * F + c], h[i]);
    }
}

// per-graph head MLP: 32 -> 16 (relu) -> 8 (relu) -> 2, one thread per graph
__global__ void k_head(const float* __restrict__ g,
                       const float* __restrict__ fw1, const float* __restrict__ fb1,
                       const float* __restrict__ fw2, const float* __restrict__ fb2,
                       const float* __restrict__ ow,  const float* __restrict__ ob,
                       float* __restrict__ out) {
    int gi = blockIdx.x * blockDim.x + threadIdx.x;
    if (gi >= N_GRAPHS) return;
    float a[32], b[16], c[8];
#pragma unroll
    for (int i = 0; i < 32; ++i) a[i] = g[(size_t)gi * 32 + i];
#pragma unroll
    for (int j = 0; j < 16; ++j) {
        float s = fb1[j];
        for (int i = 0; i < 32; ++i) s += a[i] * fw1[i * 16 + j];
        b[j] = fmaxf(s, 0.0f);
    }
#pragma unroll
    for (int j = 0; j < 8; ++j) {
        float s = fb2[j];
        for (int i = 0; i < 16; ++i) s += b[i] * fw2[i * 8 + j];
        c[j] = fmaxf(s, 0.0f);
    }
#pragma unroll
    for (int j = 0; j < 2; ++j) {
        float s = ob[j];
        for (int i = 0; i < 8; ++i) s += c[i] * ow[i * 2 + j];
        out[(size_t)gi * 2 + j] = s;
    }
}

// ------------------------------------------------------------------ launch
static inline int blocks_for(long n) {
    long b = (n + 255) / 256;
    return (int)(b > 16384 ? 16384 : (b < 1 ? 1 : b));
}

static void launch_gemm(float* out, const float* A, const float* W, const float* bias,
                        int M, int K, int N, int doRelu, hipStream_t stream) {
    long tilesM  = (M + 15) / 16;
    long groupsN = ((N + 15) / 16 + 1) / 2;
    long threads = tilesM * groupsN * 32;
    int  blocks  = (int)((threads + 255) / 256);
    k_gemm_wmma_f32<<<blocks, 256, 0, stream>>>(A, W, bias, out, M, K, N, doRelu);
}

extern "C" void kernel_launch(void* const* d_in, const int* in_sizes, int n_in,
                              void* d_out, int out_size, void* d_ws, size_t ws_size,
                              hipStream_t stream) {
    (void)in_sizes; (void)n_in; (void)out_size; (void)ws_size;

    const float* x     = (const float*)d_in[0];
    const int*   eidx  = (const int*)d_in[1];
    const int*   batch = (const int*)d_in[2];
    const float *w1a = (const float*)d_in[3],  *b1a = (const float*)d_in[4];
    const float *w1b = (const float*)d_in[5],  *b1b = (const float*)d_in[6];
    const float *g1  = (const float*)d_in[7],  *be1 = (const float*)d_in[8];
    const float *w2a = (const float*)d_in[9],  *b2a = (const float*)d_in[10];
    const float *w2b = (const float*)d_in[11], *b2b = (const float*)d_in[12];
    const float *g2  = (const float*)d_in[13], *be2 = (const float*)d_in[14];
    const float *w3a = (const float*)d_in[15], *b3a = (const float*)d_in[16];
    const float *w3b = (const float*)d_in[17], *b3b = (const float*)d_in[18];
    const float *g3  = (const float*)d_in[19], *be3 = (const float*)d_in[20];
    const float *fw1 = (const float*)d_in[21], *fb1 = (const float*)d_in[22];
    const float *fw2 = (const float*)d_in[23], *fb2 = (const float*)d_in[24];
    const float *ow  = (const float*)d_in[25], *ob  = (const float*)d_in[26];
    float* out = (float*)d_out;

    const int* src = eidx;
    const int* dst = eidx + N_EDGES;

    // workspace layout (floats)
    float* ws = (float*)d_ws;
    const size_t NT = (size_t)N_NODES * 198;
    float* T      = ws;              // pre-MLP accumulator  (N x Fin)
    float* Z      = ws + NT;         // hidden activations   (N x Fhid)
    float* H      = ws + 2 * NT;     // layer output         (N x Fout)
    float* stats  = ws + 3 * NT;     // 2*198 per-channel sums
    float* pooled = stats + 2 * 198; // 1024 x 32

    // ---------------- layer 1: 114 -> 198 -> 198
    k_copy_f32x4<<<blocks_for((long)N_NODES * 114 / 4), 256, 0, stream>>>(
        (float4*)T, (const float4*)x, N_NODES * 114u / 4u);
    k_scatter_add<114><<<blocks_for((long)N_EDGES * 57), 256, 0, stream>>>(T, x, src, dst, N_EDGES);
    launch_gemm(Z, T, w1a, b1a, N_NODES, 114, 198, 1, stream);
    launch_gemm(H, Z, w1b, b1b, N_NODES, 198, 198, 1, stream);   // outer relu fused
    k_zero_f32<<<1, 256, 0, stream>>>(stats, 2 * 198);
    k_bn_stats<198><<<1024, 256, 0, stream>>>(H, stats, N_NODES);
    k_bn_apply<198><<<blocks_for((long)N_NODES * 198), 256, 0, stream>>>(H, stats, g1, be1, N_NODES);

    // ---------------- layer 2: 198 -> 64 -> 64
    k_copy_f32x4<<<blocks_for((long)N_NODES * 198 / 4), 256, 0, stream>>>(
        (float4*)T, (const float4*)H, N_NODES * 198u / 4u);
    k_scatter_add<198><<<blocks_for((long)N_EDGES * 99), 256, 0, stream>>>(T, H, src, dst, N_EDGES);
    launch_gemm(Z, T, w2a, b2a, N_NODES, 198, 64, 1, stream);
    launch_gemm(H, Z, w2b, b2b, N_NODES, 64, 64, 1, stream);
    k_zero_f32<<<1, 256, 0, stream>>>(stats, 2 * 64);
    k_bn_stats<64><<<1024, 256, 0, stream>>>(H, stats, N_NODES);
    k_bn_apply<64><<<blocks_for((long)N_NODES * 64), 256, 0, stream>>>(H, stats, g2, be2, N_NODES);

    // ---------------- layer 3: 64 -> 32 -> 32
    k_copy_f32x4<<<blocks_for((long)N_NODES * 64 / 4), 256, 0, stream>>>(
        (float4*)T, (const float4*)H, N_NODES * 64u / 4u);
    k_scatter_add<64><<<blocks_for((long)N_EDGES * 32), 256, 0, stream>>>(T, H, src, dst, N_EDGES);
    launch_gemm(Z, T, w3a, b3a, N_NODES, 64, 32, 1, stream);
    launch_gemm(H, Z, w3b, b3b, N_NODES, 32, 32, 1, stream);
    k_zero_f32<<<1, 256, 0, stream>>>(stats, 2 * 32);
    k_bn_stats<32><<<1024, 256, 0, stream>>>(H, stats, N_NODES);
    k_bn_apply<32><<<blocks_for((long)N_NODES * 32), 256, 0, stream>>>(H, stats, g3, be3, N_NODES);

    // ---------------- pool + head
    k_zero_f32<<<blocks_for(N_GRAPHS * 32), 256, 0, stream>>>(pooled, N_GRAPHS * 32u);
    k_pool<32><<<blocks_for((long)N_NODES * 32), 256, 0, stream>>>(H, batch, pooled, N_NODES);
    k_head<<<(N_GRAPHS + 255) / 256, 256, 0, stream>>>(pooled, fw1, fb1, fw2, fb2, ow, ob, out);
}